// DNA_7748121002705
// MI455X (gfx1250) — compile-verified
//
#include <hip/hip_runtime.h>
#include <hip/hip_bf16.h>
#include <math.h>

typedef __attribute__((ext_vector_type(2))) float v2f;
typedef __attribute__((ext_vector_type(4))) float v4f;
typedef __attribute__((ext_vector_type(8))) float v8f;

#define CCH   8      // channels
#define LMAX  4      // x_all slots: input + 3 layers
#define NHEAD 2
#define DHEAD 4

// ---------------------------------------------------------------- degree ----
__global__ void deg_init_kernel(float* deg, int n) {
  int i = blockIdx.x * blockDim.x + threadIdx.x;
  if (i < n) deg[i] = 1.0f;  // self loop contributes 1
}

__global__ void deg_accum_kernel(const int* __restrict__ dst, float* deg, int e) {
  int i = blockIdx.x * blockDim.x + threadIdx.x;
  if (i < e) atomicAdd(&deg[dst[i]], 1.0f);
}

__global__ void deg_fin_kernel(float* deg, int n) {
  int i = blockIdx.x * blockDim.x + threadIdx.x;
  if (i < n) {
    float d = deg[i];
    deg[i] = d > 0.0f ? rsqrtf(d) : 0.0f;
  }
}

__global__ void zero_kernel(float* p, int n) {
  int i = blockIdx.x * blockDim.x + threadIdx.x;
  if (i < n) p[i] = 0.0f;
}

// ------------------------------------------------- input projection (WMMA) --
// h0 = relu([emb[node_index] | x] @ W1 + b1), stored into x_all slot 0.
// One wave per 16-node tile: D(16x16) = A(16x24) * B(24x16), B = W1 zero-
// padded to 16 columns; K split into 6 steps of 4 (v_wmma_f32_16x16x4_f32).
// A/B fragment loads are branchless (pointer select + masked multiply) so no
// exec-mask churn surrounds the WMMA chain.
__global__ void input_proj_kernel(const float* __restrict__ x,
                                  const int*   __restrict__ node_index,
                                  const float* __restrict__ emb,
                                  const float* __restrict__ W1,
                                  const float* __restrict__ b1,
                                  float* __restrict__ x_all, int N) {
  const int lane = threadIdx.x & 31;
  const int wave = threadIdx.x >> 5;
  const int base = (blockIdx.x * (blockDim.x >> 5) + wave) * 16;
  if (base >= N) return;  // wave-uniform exit: EXEC stays all-ones for WMMA

  const int m  = lane & 15;            // A row (node within tile)
  const int kb = (lane >> 4) << 1;     // K sub-offset per A/B layout (0 or 2)
  const int n  = lane & 15;            // B/D column
  const int node = base + m;
  const int ld   = node < N ? node : N - 1;   // clamped load row
  const int er   = node_index[ld];            // embedding row

  const float* embrow = emb + (size_t)er * 8;
  const float* xrow   = x   + (size_t)ld * 16 - 8;   // so xrow[k0] = x[ld][k0-8]
  const int    nc     = n & 7;                       // clamped B column
  const float  bmask  = (n < CCH) ? 1.0f : 0.0f;

  v8f acc = {};
#pragma unroll
  for (int ks = 0; ks < 6; ++ks) {
    const int k0 = ks * 4 + kb;   // even; k0 and k0+1 always in same region
    const float* pa = (k0 < 8) ? (embrow + k0) : (xrow + k0);
    v2f a = *(const v2f*)pa;      // 8B-aligned: k0 even in both regions
    v2f b;
    b.x = W1[(size_t)k0 * CCH + nc] * bmask;
    b.y = W1[(size_t)(k0 + 1) * CCH + nc] * bmask;
    acc = __builtin_amdgcn_wmma_f32_16x16x4_f32(
        false, a, false, b, (short)0, acc, false, false);
  }

  if (n < CCH) {
    const float bias = b1[n];
#pragma unroll
    for (int r = 0; r < 8; ++r) {
      const int row = r + ((lane >> 4) << 3);   // C/D layout: VGPR r -> M=r or r+8
      const int nd  = base + row;
      if (nd < N) {
        float v = acc[r] + bias;
        x_all[((size_t)nd * LMAX + 0) * CCH + n] = v > 0.0f ? v : 0.0f;
      }
    }
  }
}

// ------------------------------------------------------- edge attention -----
// One thread per (edge or self-loop). Diagonal q/k/v (grouped 1x1 linear)
// applied on the fly; all x_all row gathers are 2x global_load_b128 (rows are
// 32B-aligned in the workspace).
template <int L>
__global__ void edge_attn_kernel(const int*   __restrict__ src,
                                 const int*   __restrict__ dst,
                                 const float* __restrict__ dinv,
                                 const float* __restrict__ x_all,
                                 float*       __restrict__ agg,
                                 const float* __restrict__ Wq, const float* __restrict__ bq,
                                 const float* __restrict__ Wk, const float* __restrict__ bk,
                                 const float* __restrict__ Wv, const float* __restrict__ bv,
                                 int E, int N) {
  int e = blockIdx.x * blockDim.x + threadIdx.x;
  if (e >= E + N) return;
  int s, t;
  if (e < E) { s = src[e]; t = dst[e]; }
  else       { s = t = e - E; }                 // self loop
  const float nrm = dinv[s] * dinv[t];

  float wq[CCH], cbq[CCH], wk[CCH], cbk[CCH], wv[CCH], cbv[CCH];
#pragma unroll
  for (int c = 0; c < CCH; ++c) {
    wq[c] = Wq[c]; cbq[c] = bq[c];
    wk[c] = Wk[c]; cbk[c] = bk[c];
    wv[c] = Wv[c]; cbv[c] = bv[c];
  }

  // q = diag(Wq) * x_all[t, L-1, :] + bq   (two b128 gathers)
  float q[CCH];
  {
    const v4f* pq = (const v4f*)(x_all + ((size_t)t * LMAX + (L - 1)) * CCH);
    v4f qlo = pq[0], qhi = pq[1];
#pragma unroll
    for (int c = 0; c < 4; ++c) {
      q[c]     = qlo[c] * wq[c]     + cbq[c];
      q[c + 4] = qhi[c] * wq[c + 4] + cbq[c + 4];
    }
  }

  // gather all L source rows (2 b128 each)
  float xs[L][CCH];
#pragma unroll
  for (int l = 0; l < L; ++l) {
    const v4f* p = (const v4f*)(x_all + ((size_t)s * LMAX + l) * CCH);
    v4f lo = p[0], hi = p[1];
#pragma unroll
    for (int c = 0; c < 4; ++c) { xs[l][c] = lo[c]; xs[l][c + 4] = hi[c]; }
  }

  float outv[CCH];
#pragma unroll
  for (int h = 0; h < NHEAD; ++h) {
    float sc[L];
#pragma unroll
    for (int l = 0; l < L; ++l) {
      float a = 0.0f;
#pragma unroll
      for (int d = 0; d < DHEAD; ++d) {
        const int c = h * DHEAD + d;
        a += q[c] * (xs[l][c] * wk[c] + cbk[c]);
      }
      sc[l] = a * 0.5f;   // 1/sqrt(d), d=4
    }
    float mx = sc[0];
#pragma unroll
    for (int l = 1; l < L; ++l) mx = fmaxf(mx, sc[l]);
    float ssum = 0.0f;
#pragma unroll
    for (int l = 0; l < L; ++l) { sc[l] = expf(sc[l] - mx); ssum += sc[l]; }
    const float inv = 1.0f / ssum;
#pragma unroll
    for (int d = 0; d < DHEAD; ++d) {
      const int c = h * DHEAD + d;
      float a = 0.0f;
#pragma unroll
      for (int l = 0; l < L; ++l) a += sc[l] * (xs[l][c] * wv[c] + cbv[c]);
      outv[c] = a * inv;
    }
  }

#pragma unroll
  for (int c = 0; c < CCH; ++c)
    atomicAdd(&agg[(size_t)t * CCH + c], outv[c] * nrm);
}

__global__ void layer_fin_kernel(const float* __restrict__ agg,
                                 float* __restrict__ x_all, int N, int L) {
  int i = blockIdx.x * blockDim.x + threadIdx.x;
  if (i >= N * CCH) return;
  const int n = i >> 3, c = i & 7;
  const float v = agg[i];
  x_all[((size_t)n * LMAX + L) * CCH + c] = v > 0.0f ? v : 0.0f;
}

// ---------------------------------------------------------- final logits ----
__global__ void logits_kernel(const float* __restrict__ x_all,
                              const float* __restrict__ W2,
                              const float* __restrict__ b2,
                              float* __restrict__ out, int N) {
  int n = blockIdx.x * blockDim.x + threadIdx.x;
  if (n >= N) return;
  const v4f* ph = (const v4f*)(x_all + ((size_t)n * LMAX + 3) * CCH);
  v4f hlo = ph[0], hhi = ph[1];
  float l0 = b2[0], l1 = b2[1];
#pragma unroll
  for (int c = 0; c < 4; ++c) {
    l0 += hlo[c] * W2[c * 2 + 0] + hhi[c] * W2[(c + 4) * 2 + 0];
    l1 += hlo[c] * W2[c * 2 + 1] + hhi[c] * W2[(c + 4) * 2 + 1];
  }
  const float m   = fmaxf(l0, l1);
  const float lse = m + logf(expf(l0 - m) + expf(l1 - m));
  out[(size_t)n * 2 + 0] = l0 - lse;
  out[(size_t)n * 2 + 1] = l1 - lse;
}

// --------------------------------------------------------------- launcher ---
extern "C" void kernel_launch(void* const* d_in, const int* in_sizes, int n_in,
                              void* d_out, int out_size, void* d_ws, size_t ws_size,
                              hipStream_t stream) {
  const float* x          = (const float*)d_in[0];
  const int*   node_index = (const int*)  d_in[1];
  // d_in[2] node_one_hot: unused by reference
  const int*   edge_index = (const int*)  d_in[3];
  // d_in[4] edge_weight: unused by reference (gcn_norm uses ones)
  const float* emb        = (const float*)d_in[5];
  const float* W1         = (const float*)d_in[6];
  const float* b1         = (const float*)d_in[7];
  const float* Wq         = (const float*)d_in[8];
  const float* bq         = (const float*)d_in[9];
  const float* Wk         = (const float*)d_in[10];
  const float* bk         = (const float*)d_in[11];
  const float* Wv         = (const float*)d_in[12];
  const float* bv         = (const float*)d_in[13];
  const float* W2         = (const float*)d_in[14];
  const float* b2         = (const float*)d_in[15];

  const int N = in_sizes[0] / 16;   // FEAT = 16
  const int E = in_sizes[3] / 2;    // edge_index is (2, E)
  const int* src = edge_index;
  const int* dst = edge_index + E;

  // Workspace layout: dinv[N] | x_all[N][4][8] | agg[N][8]
  // (d_ws is 256B-aligned; N*4 and N*132 byte offsets keep 16B alignment
  //  for the b128 row gathers as long as N is a multiple of 4 — N=100000.)
  float* dinv  = (float*)d_ws;
  float* x_all = dinv + N;
  float* agg   = x_all + (size_t)N * LMAX * CCH;
  float* out   = (float*)d_out;

  const int T = 256;
  const int gN  = (N + T - 1) / T;
  const int gE  = (E + T - 1) / T;
  const int gEN = (E + N + T - 1) / T;
  const int gNC = (N * CCH + T - 1) / T;
  const int gTile = (N + 16 * (T / 32) - 1) / (16 * (T / 32));  // 16 nodes/wave

  // 1) symmetric GCN normalization (degree includes self loop)
  deg_init_kernel<<<gN, T, 0, stream>>>(dinv, N);
  deg_accum_kernel<<<gE, T, 0, stream>>>(dst, dinv, E);
  deg_fin_kernel<<<gN, T, 0, stream>>>(dinv, N);

  // 2) input projection via v_wmma_f32_16x16x4_f32 -> x_all slot 0
  input_proj_kernel<<<gTile, T, 0, stream>>>(x, node_index, emb, W1, b1, x_all, N);

  // 3) three DNA attention layers
  for (int i = 0; i < 3; ++i) {
    zero_kernel<<<gNC, T, 0, stream>>>(agg, N * CCH);
    const float* wqi = Wq + (size_t)i * CCH; const float* bqi = bq + (size_t)i * CCH;
    const float* wki = Wk + (size_t)i * CCH; const float* bki = bk + (size_t)i * CCH;
    const float* wvi = Wv + (size_t)i * CCH; const float* bvi = bv + (size_t)i * CCH;
    switch (i + 1) {
      case 1: edge_attn_kernel<1><<<gEN, T, 0, stream>>>(src, dst, dinv, x_all, agg,
                  wqi, bqi, wki, bki, wvi, bvi, E, N); break;
      case 2: edge_attn_kernel<2><<<gEN, T, 0, stream>>>(src, dst, dinv, x_all, agg,
                  wqi, bqi, wki, bki, wvi, bvi, E, N); break;
      default: edge_attn_kernel<3><<<gEN, T, 0, stream>>>(src, dst, dinv, x_all, agg,
                  wqi, bqi, wki, bki, wvi, bvi, E, N); break;
    }
    layer_fin_kernel<<<gNC, T, 0, stream>>>(agg, x_all, N, i + 1);
  }

  // 4) classifier + log_softmax
  logits_kernel<<<gN, T, 0, stream>>>(x_all, W2, b2, out, N);
}